// GConv_48369921687842
// MI455X (gfx1250) — compile-verified
//
#include <hip/hip_runtime.h>
#include <stdint.h>

#ifndef __has_builtin
#define __has_builtin(x) 0
#endif

#define GC_B   4
#define GC_C   64
#define GC_BC  256      // B*C
#define GC_N1  65536
#define GC_N2  65536
#define GC_K   9
#define GC_F   6
#define GC_TN  16       // n-tile per block (row = 64 B in the out tile)
#define GC_PAD 20       // LDS row stride (dwords): 16B-aligned rows, 2-way max bank conflict
#define GC_CPR (GC_TN / 4)                       // 16B chunks per row = 4
#define GC_NCH (GC_F * GC_BC * GC_CPR)           // chunks per block = 6144

// gfx12 CPol: TH[2:0]; store TH=1 -> NT (non-temporal streaming store)
#define GC_CPOL_NT 1

typedef float v4f __attribute__((ext_vector_type(4)));
typedef int   v4i __attribute__((ext_vector_type(4)));
typedef __attribute__((address_space(1))) v4i glb_v4i;   // global b128 chunk
typedef __attribute__((address_space(3))) v4i lds_v4i;   // LDS b128 chunk

// ---------------------------------------------------------------------------
// Kernel 1: transpose x[bc][j]  (256 x 65536)  ->  xT[j][bc]  (65536 x 256)
// xT (67 MB) stays resident in the 192 MB L2; it turns the per-(n,k) gather
// into a fully-coalesced contiguous 1 KB row read. x itself is read once (NT).
// ---------------------------------------------------------------------------
__global__ __launch_bounds__(256) void gc_xpose(const float* __restrict__ x,
                                                float* __restrict__ xT) {
  __shared__ float tile[32][33];
  const int j0  = blockIdx.x << 5;
  const int bc0 = blockIdx.y << 5;
  const int tx  = threadIdx.x & 31;
  const int ty0 = threadIdx.x >> 5;
#pragma unroll
  for (int p = 0; p < 4; ++p) {
    const int ty = ty0 + (p << 3);
    tile[ty][tx] = __builtin_nontemporal_load(
        x + (size_t)(bc0 + ty) * GC_N1 + (j0 + tx));
  }
  __syncthreads();
#pragma unroll
  for (int p = 0; p < 4; ++p) {
    const int ty = ty0 + (p << 3);
    xT[(size_t)(j0 + ty) * GC_BC + (bc0 + tx)] = tile[tx][ty];  // RT: keep in L2
  }
}

// ---------------------------------------------------------------------------
// Kernel 2: main compute. lane == bc (0..255), block covers GC_TN n-values.
//  - idx / w indexed by uniform (n,f,k) -> scalar SMEM loads (s_load)
//  - gather rows xT[j][lane] -> coalesced global_load_b32 (RT: L2-resident)
//  - acc staged in padded LDS tile; drained via CDNA5 async LDS->global b128
//    (NT policy) with chunk-within-row as the fast lane index -> 64 B segments
// ---------------------------------------------------------------------------
__global__ __launch_bounds__(256) void gc_main(const float* __restrict__ xT,
                                               const float* __restrict__ fw,
                                               const int*   __restrict__ idx,
                                               float* __restrict__ out) {
  __shared__ float otile[GC_F * GC_BC * GC_PAD];  // 6*256*20*4 = 120 KB
  const int lane = threadIdx.x;                   // == bc
  const int n0   = blockIdx.x * GC_TN;

#pragma unroll 2
  for (int i = 0; i < GC_TN; ++i) {
    const int n = n0 + i;
    float v[GC_K];
#pragma unroll
    for (int k = 0; k < GC_K; ++k) {
      const int j = idx[n * GC_K + k];            // uniform -> s_load
      v[k] = xT[((size_t)j << 8) + lane];         // coalesced 1KB row gather
    }
#pragma unroll
    for (int f = 0; f < GC_F; ++f) {
      const float* wp = fw + ((size_t)f * GC_N2 + n) * GC_K;  // uniform -> s_load
      float acc = 0.f;
#pragma unroll
      for (int k = 0; k < GC_K; ++k) acc = __builtin_fmaf(wp[k], v[k], acc);
      otile[(f * GC_BC + lane) * GC_PAD + i] = acc;
    }
  }
  __syncthreads();

  // Drain: 1536 rows (f,bc) of GC_TN floats; 4 x b128 chunks per row.
  // cid -> (row = cid>>2, q = cid&3): consecutive lanes cover consecutive
  // chunks of the same row first -> 64 B contiguous global segments.
#pragma unroll
  for (int t = 0; t < GC_NCH / 256; ++t) {        // 24 iterations
    const int cid = t * 256 + lane;
    const int q   = cid & (GC_CPR - 1);
    const int row = cid >> 2;                     // = f*256 + bc
    const int f   = row >> 8;
    const int bc  = row & 255;
    float* dst    = out + ((size_t)bc * GC_F + f) * GC_N2 + n0 + q * 4;
    const int lo  = row * GC_PAD + q * 4;         // dwords; 16B-aligned
#if __has_builtin(__builtin_amdgcn_global_store_async_from_lds_b128)
    __builtin_amdgcn_global_store_async_from_lds_b128(
        (glb_v4i*)dst, (lds_v4i*)(otile + lo), 0, GC_CPOL_NT);
#else
    __builtin_nontemporal_store(*(const v4f*)(otile + lo), (v4f*)dst);
#endif
  }
#if __has_builtin(__builtin_amdgcn_global_store_async_from_lds_b128)
# if __has_builtin(__builtin_amdgcn_s_wait_asynccnt)
  __builtin_amdgcn_s_wait_asynccnt(0);
# else
  asm volatile("s_wait_asynccnt 0" ::: "memory");
# endif
#endif
}

// ---------------------------------------------------------------------------
// Fallback (workspace too small for the 67 MB transpose): direct random gather.
// Output stays coalesced (lane == n); gathers are per-lane random.
// ---------------------------------------------------------------------------
__global__ __launch_bounds__(256) void gc_direct(const float* __restrict__ x,
                                                 const float* __restrict__ fw,
                                                 const int*   __restrict__ idx,
                                                 float* __restrict__ out) {
  const int n  = blockIdx.x * blockDim.x + threadIdx.x;
  const int bc = blockIdx.y;
  const float* xr = x + (size_t)bc * GC_N1;
  float v[GC_K];
#pragma unroll
  for (int k = 0; k < GC_K; ++k) v[k] = xr[idx[n * GC_K + k]];
#pragma unroll
  for (int f = 0; f < GC_F; ++f) {
    const float* wp = fw + ((size_t)f * GC_N2 + n) * GC_K;
    float acc = 0.f;
#pragma unroll
    for (int k = 0; k < GC_K; ++k) acc = __builtin_fmaf(wp[k], v[k], acc);
    __builtin_nontemporal_store(acc, out + ((size_t)bc * GC_F + f) * GC_N2 + n);
  }
}

extern "C" void kernel_launch(void* const* d_in, const int* in_sizes, int n_in,
                              void* d_out, int out_size, void* d_ws, size_t ws_size,
                              hipStream_t stream) {
  (void)in_sizes; (void)n_in; (void)out_size;
  const float* x   = (const float*)d_in[0];
  const float* fw  = (const float*)d_in[1];
  const int*   idx = (const int*)d_in[2];
  float* out = (float*)d_out;

  const size_t need = (size_t)GC_N1 * GC_BC * sizeof(float);  // 64 MiB
  if (ws_size >= need) {
    float* xT = (float*)d_ws;
    gc_xpose<<<dim3(GC_N1 / 32, GC_BC / 32), dim3(256), 0, stream>>>(x, xT);
    gc_main<<<dim3(GC_N2 / GC_TN), dim3(256), 0, stream>>>(xT, fw, idx, out);
  } else {
    gc_direct<<<dim3(GC_N2 / 256, GC_BC), dim3(256), 0, stream>>>(x, fw, idx, out);
  }
}